// Correlation_Loss_52415780881008
// MI455X (gfx1250) — compile-verified
//
#include <hip/hip_runtime.h>

typedef __attribute__((ext_vector_type(2))) float v2f;
typedef __attribute__((ext_vector_type(8))) float v8f;

#define PXS 256        // pixels per 16x16 patch
#define KM 32          // msi channels used (64/2)
#define KH 4           // he channels padded 3 -> 4
#define XM_STRIDE 34   // LDS row stride (floats), conflict-free + 8B aligned pairs
#define IMG 512

__global__ void corr_zero_out(float* out) { out[0] = 0.0f; }

__global__ __launch_bounds__(256)
void corr_loss_kernel(const float* __restrict__ msi,
                      const float* __restrict__ he,
                      const int* __restrict__ i_idx,
                      const int* __restrict__ j_idx,
                      float* __restrict__ out)
{
    __shared__ float XmT[PXS * XM_STRIDE];  // transposed msi patch [pixel][k]
    __shared__ float XhPos[PXS * KH];       // transposed he patch, k=3 zero-padded
    __shared__ float XhNeg[PXS * KH];       // negated copy (A operand -> Gm - Gh fused)
    __shared__ float sdiff[PXS];            // sq_m[p] - sq_h[p]
    __shared__ float smask[PXS];            // mask as 0/1 float
    __shared__ float sred[256];

    const int t  = threadIdx.x;
    const int b  = blockIdx.x;
    const int i0 = i_idx[b];
    const int j0 = j_idx[b];

    // ---- stage msi patch (32 x 256), transposed into LDS ----
    for (int idx = t; idx < KM * PXS; idx += 256) {
        int k = idx >> 8;
        int p = idx & (PXS - 1);
        int py = p >> 4, px = p & 15;
        float v = msi[((size_t)k * IMG + (size_t)(i0 + py)) * IMG + (size_t)(j0 + px)];
        XmT[p * XM_STRIDE + k] = v;
    }
    // ---- stage he patch (3 x 256), pad K to 4 with zeros, plus negated copy ----
    for (int idx = t; idx < KH * PXS; idx += 256) {
        int k = idx >> 8;
        int p = idx & (PXS - 1);
        int py = p >> 4, px = p & 15;
        float v = (k < 3)
            ? he[((size_t)k * IMG + (size_t)(i0 + py)) * IMG + (size_t)(j0 + px)]
            : 0.0f;
        XhPos[p * KH + k] = v;
        XhNeg[p * KH + k] = -v;
    }
    __syncthreads();

    // ---- per-pixel: s = |m|^2 - |h|^2 and threshold mask ----
    {
        int p = t;
        float sm = 0.f;
        #pragma unroll
        for (int k = 0; k < KM; ++k) { float v = XmT[p * XM_STRIDE + k]; sm += v * v; }
        float sh = 0.f, hs = 0.f;
        #pragma unroll
        for (int k = 0; k < 3; ++k) { float v = XhPos[p * KH + k]; sh += v * v; hs += v; }
        sdiff[p] = sm - sh;
        smask[p] = (hs >= 0.05f) ? 1.0f : 0.0f;
    }
    __syncthreads();

    // ---- WMMA fragment addressing (ISA 7.12.2: 32-bit A 16x4 layout) ----
    const int lane   = t & 31;
    const int wave   = t >> 5;
    const int m      = lane & 15;          // M (A) / N (B) within tile
    const int koff   = (lane >> 4) << 1;   // lanes 16-31 hold K+2
    const int rowoff = (lane >> 4) << 3;   // C/D: lanes 16-31 hold M+8

    float lsum = 0.0f;

    // 256 output tiles (16x16 grid of 16x16), 32 tiles per wave, uniform EXEC
    for (int n = 0; n < 32; ++n) {
        int tile  = wave * 32 + n;
        int pbase = (tile >> 4) << 4;
        int qbase = (tile & 15) << 4;

        v8f acc = {};
        // Gm accumulation: 8 x v_wmma_f32_16x16x4_f32 over K=32
        #pragma unroll
        for (int k0 = 0; k0 < KM; k0 += 4) {
            v2f av = *(const v2f*)&XmT[(pbase + m) * XM_STRIDE + k0 + koff];
            v2f bv = *(const v2f*)&XmT[(qbase + m) * XM_STRIDE + k0 + koff];
            acc = __builtin_amdgcn_wmma_f32_16x16x4_f32(
                false, av, false, bv, (short)0, acc, false, false);
        }
        // minus Gh in the same accumulator: (-Xh)^T * Xh, K padded to 4
        {
            v2f av = *(const v2f*)&XhNeg[(pbase + m) * KH + koff];
            v2f bv = *(const v2f*)&XhPos[(qbase + m) * KH + koff];
            acc = __builtin_amdgcn_wmma_f32_16x16x4_f32(
                false, av, false, bv, (short)0, acc, false, false);
        }

        // epilogue: |s[p] + s[q] - 2*(Gm-Gh)| * mask[p] * mask[q]
        int   q    = qbase + m;
        float sq_q = sdiff[q];
        float mq   = smask[q];
        #pragma unroll
        for (int r = 0; r < 8; ++r) {
            int p = pbase + r + rowoff;
            lsum += fabsf(sdiff[p] + sq_q - 2.0f * acc[r]) * smask[p] * mq;
        }
    }

    // ---- block reduction, then one atomic per patch ----
    sred[t] = lsum;
    __syncthreads();
    for (int off = 128; off > 0; off >>= 1) {
        if (t < off) sred[t] += sred[t + off];
        __syncthreads();
    }
    if (t == 0) {
        // mean over 256*256 elements, then sum / (batch//5 = 160)
        atomicAdd(out, sred[0] * (1.0f / (65536.0f * 160.0f)));
    }
}

extern "C" void kernel_launch(void* const* d_in, const int* in_sizes, int n_in,
                              void* d_out, int out_size, void* d_ws, size_t ws_size,
                              hipStream_t stream)
{
    const float* msi = (const float*)d_in[0];
    const float* he  = (const float*)d_in[1];
    const int*   ii  = (const int*)d_in[2];
    const int*   jj  = (const int*)d_in[3];
    float*       out = (float*)d_out;

    const int nb = in_sizes[2];  // 800 patches

    corr_zero_out<<<1, 1, 0, stream>>>(out);
    corr_loss_kernel<<<nb, 256, 0, stream>>>(msi, he, ii, jj, out);
}